// DynamicDenseBlock_57492432224963
// MI455X (gfx1250) — compile-verified
//
#include <hip/hip_runtime.h>

// ---------------------------------------------------------------------------
// Fused: RMSNorm -> x @ W1^T -> GELU(exact) -> @ W2^T -> permute (C,B,T,L)
// x[4,4096,4096] f32, W1[48,4096], W2[48,48], out[4,4,4096,12]
//
// HBM-bound (256 MB x-read, AI~25 flop/B -> needs ~580 TF/s fp32 => WMMA).
// v1 used per-lane strided fragment loads from global: 16 cache lines per
// load = request fragmentation that caps streaming bandwidth. v2 stages x
// tiles + W1 K-chunks into LDS via GLOBAL_LOAD_ASYNC_TO_LDS_B128 (coalesced
// 512B row segments, ASYNCcnt-tracked, double buffered), and reads WMMA
// fragments from LDS with a bank-conflict-free pitch (KC+4 floats:
// bank = 4r + k + 2h covers all 64 banks across the wave).
// ---------------------------------------------------------------------------

typedef __attribute__((ext_vector_type(2))) float v2f;
typedef __attribute__((ext_vector_type(8))) float v8f;

#define DDIM   4096
#define LTOT   48
#define TDIM   4096
#define NB     4
#define EPSF   1e-5f
#define KSPLIT 2048                 // K range per wave (DDIM/2)
#define KC     128                  // K chunk staged per double-buffer step
#define NCHUNK (KSPLIT / KC)        // 16
#define XPITCH (KC + 4)             // 132 floats: conflict-free ds_load_b64
#define XCHB   (16 * XPITCH * 4)    // 8448 B  : one x tile chunk
#define W1CHB  (48 * XPITCH * 4)    // 25344 B : one W1 chunk
#define GPITCH 50

// dynamic-LDS layout (bytes)
#define XBUF_OFF  0u                                   // 8 waves x 2 bufs
#define W1BUF_OFF (XBUF_OFF + 8u * 2u * XCHB)          // 2 slices x 2 bufs
#define PART_OFF  (W1BUF_OFF + 2u * 2u * W1CHB)        // 4 x 32 x 25 f32
#define GSH_OFF   (PART_OFF + 4u * 32u * 25u * 4u)     // 4 x 16 x 50 f32
#define LDS_TOTAL (GSH_OFF + 4u * 16u * GPITCH * 4u)   // = 262144 B

// one async DMA: 32 lanes x 16B = one contiguous 512B row segment -> LDS
__device__ __forceinline__ void stage_row_async(unsigned lds_off,
                                                const float* gsrc, int lane) {
  unsigned ldsa = lds_off + (unsigned)lane * 16u;
  unsigned voff = (unsigned)lane * 16u;
  asm volatile("global_load_async_to_lds_b128 %0, %1, %2"
               :: "v"(ldsa), "v"(voff), "s"(gsrc) : "memory");
}

__global__ __launch_bounds__(256)
void ddb_fused_kernel(const float* __restrict__ x,
                      const float* __restrict__ W1,
                      const float* __restrict__ W2,
                      float* __restrict__ out) {
  extern __shared__ char smem[];

  const int lane   = threadIdx.x & 31;
  const int wave   = threadIdx.x >> 5;
  const int tslot  = wave >> 1;                 // tile slot in block, 0..3
  const int kslice = wave & 1;                  // which K half
  const int tile   = blockIdx.x * 4 + tslot;    // 0..1023
  const int rowbase = tile * 16;

  const int id16 = lane & 15;                   // fragment row / col
  const int half = lane >> 4;                   // K-pair select (ISA 7.12.2)
  const int koff = half * 2;

  const unsigned xb_base  = XBUF_OFF + (unsigned)(wave * 2) * XCHB;
  const unsigned w1b_base = W1BUF_OFF + (unsigned)(kslice * 2) * W1CHB;
  const int kbeg = kslice * KSPLIT;

  // ---- issue async staging for one K chunk --------------------------------
  auto issue_chunk = [&](int ci) {
    const int kc = kbeg + ci * KC;
    const unsigned xb  = xb_base + (unsigned)(ci & 1) * XCHB;
    const unsigned w1b = w1b_base + (unsigned)(ci & 1) * W1CHB;
#pragma unroll
    for (int r = 0; r < 16; ++r)                       // own x tile rows
      stage_row_async(xb + (unsigned)r * (XPITCH * 4),
                      x + (size_t)(rowbase + r) * DDIM + kc, lane);
#pragma unroll
    for (int r = 0; r < 12; ++r) {                     // 12 of 48 W1 rows
      const int row = tslot * 12 + r;
      stage_row_async(w1b + (unsigned)row * (XPITCH * 4),
                      W1 + (size_t)row * DDIM + kc, lane);
    }
  };

  v8f c0 = {}; v8f c1 = {}; v8f c2 = {};
  float ss0 = 0.0f, ss1 = 0.0f;

  issue_chunk(0);                                // prologue: 28 async ops

  for (int i = 0; i < NCHUNK; ++i) {
    if (i + 1 < NCHUNK) {
      issue_chunk(i + 1);                        // overlap next chunk DMA
      asm volatile("s_wait_asynccnt 28" ::: "memory");   // chunk i complete
    } else {
      asm volatile("s_wait_asynccnt 0" ::: "memory");
    }
    __syncthreads();                             // W1 staged by peer waves

    if (i + 2 < NCHUNK) {                        // prefetch 2 chunks ahead
      const float* pf = x + (size_t)(rowbase + id16) * DDIM
                          + (kbeg + (i + 2) * KC) + half * 64;
      __builtin_prefetch(pf, 0, 0);              // lines {0,2} of each row
      __builtin_prefetch(pf + 32, 0, 0);         // lines {1,3}
    }

    const float* xt = (const float*)(smem + xb_base + (unsigned)(i & 1) * XCHB);
    const float* wt = (const float*)(smem + w1b_base + (unsigned)(i & 1) * W1CHB);
    const float* af  = xt + id16 * XPITCH + koff;
    const float* b0f = wt + (0 * 16 + id16) * XPITCH + koff;
    const float* b1f = wt + (1 * 16 + id16) * XPITCH + koff;
    const float* b2f = wt + (2 * 16 + id16) * XPITCH + koff;

#pragma unroll 4
    for (int k = 0; k < KC; k += 4) {
      v2f a  = *(const v2f*)(af + k);            // ds_load_b64, conflict-free
      v2f b0 = *(const v2f*)(b0f + k);
      v2f b1 = *(const v2f*)(b1f + k);
      v2f b2 = *(const v2f*)(b2f + k);
      ss0 = fmaf(a.x, a.x, ss0);
      ss1 = fmaf(a.y, a.y, ss1);
      c0 = __builtin_amdgcn_wmma_f32_16x16x4_f32(false, a, false, b0, (short)0, c0, false, false);
      c1 = __builtin_amdgcn_wmma_f32_16x16x4_f32(false, a, false, b1, (short)0, c1, false, false);
      c2 = __builtin_amdgcn_wmma_f32_16x16x4_f32(false, a, false, b2, (short)0, c2, false, false);
    }
    __syncthreads();           // chunk i fully consumed before buf reuse
  }

  float ss = ss0 + ss1;
  float* part = (float*)(smem + PART_OFF);       // [4][32][25]
  float* gsh  = (float*)(smem + GSH_OFF);        // [4][16][GPITCH]

  // ---- cross-wave K reduction ---------------------------------------------
  if (kslice == 1) {
    float* p = part + (tslot * 32 + lane) * 25;
#pragma unroll
    for (int v = 0; v < 8; ++v) { p[v] = c0[v]; p[8 + v] = c1[v]; p[16 + v] = c2[v]; }
    p[24] = ss;
  }
  __syncthreads();

  if (kslice == 0) {
    const float* p = part + (tslot * 32 + lane) * 25;
#pragma unroll
    for (int v = 0; v < 8; ++v) { c0[v] += p[v]; c1[v] += p[8 + v]; c2[v] += p[16 + v]; }
    ss += p[24];

    ss += __shfl_xor(ss, 16, 32);                // join the two K-pair halves
    const float scale = rsqrtf(ss * (1.0f / (float)DDIM) + EPSF);

    // RMS scale + exact GELU; stage to LDS converting C-layout -> rows
#pragma unroll
    for (int v = 0; v < 8; ++v) {
      const int r = half ? (v + 8) : v;
      const float sv = __shfl(scale, r, 32);     // row r scale lives in lane r
      float h0 = c0[v] * sv, h1 = c1[v] * sv, h2 = c2[v] * sv;
      float* gr = gsh + (tslot * 16 + r) * GPITCH;
      gr[0 * 16 + id16] = 0.5f * h0 * (1.0f + erff(h0 * 0.70710678118654752f));
      gr[1 * 16 + id16] = 0.5f * h1 * (1.0f + erff(h1 * 0.70710678118654752f));
      gr[2 * 16 + id16] = 0.5f * h2 * (1.0f + erff(h2 * 0.70710678118654752f));
    }
  }
  __syncthreads();

  if (kslice == 0) {
    // ---- GEMM2: dw[16x48] = g[16x48] @ W2^T, via WMMA ---------------------
    v8f d0 = {}; v8f d1 = {}; v8f d2 = {};
    const float* ga   = gsh + (tslot * 16 + id16) * GPITCH + koff;
    const float* w2p0 = W2 + (size_t)(0 * 16 + id16) * LTOT + koff;
    const float* w2p1 = W2 + (size_t)(1 * 16 + id16) * LTOT + koff;
    const float* w2p2 = W2 + (size_t)(2 * 16 + id16) * LTOT + koff;
#pragma unroll
    for (int k = 0; k < LTOT; k += 4) {
      v2f a  = *(const v2f*)(ga + k);
      v2f b0 = *(const v2f*)(w2p0 + k);
      v2f b1 = *(const v2f*)(w2p1 + k);
      v2f b2 = *(const v2f*)(w2p2 + k);
      d0 = __builtin_amdgcn_wmma_f32_16x16x4_f32(false, a, false, b0, (short)0, d0, false, false);
      d1 = __builtin_amdgcn_wmma_f32_16x16x4_f32(false, a, false, b1, (short)0, d1, false, false);
      d2 = __builtin_amdgcn_wmma_f32_16x16x4_f32(false, a, false, b2, (short)0, d2, false, false);
    }

    // ---- permuted store: out[c][b][t][j] = dw[b,t, c*12+j] ----------------
#pragma unroll
    for (int v = 0; v < 8; ++v) {
      const int grow = rowbase + (half ? (v + 8) : v);
      const int bi = grow >> 12;
      const int ti = grow & (TDIM - 1);
#pragma unroll
      for (int nt = 0; nt < 3; ++nt) {
        const int l = nt * 16 + id16;
        const int ci = l / 12;
        const int j  = l - ci * 12;
        const float dv = (nt == 0) ? d0[v] : ((nt == 1) ? d1[v] : d2[v]);
        out[((size_t)(ci * NB + bi) * TDIM + ti) * 12 + j] = dv;
      }
    }
  }
}

extern "C" void kernel_launch(void* const* d_in, const int* in_sizes, int n_in,
                              void* d_out, int out_size, void* d_ws, size_t ws_size,
                              hipStream_t stream) {
  (void)in_sizes; (void)n_in; (void)d_ws; (void)ws_size; (void)out_size;
  const float* x  = (const float*)d_in[0];
  const float* W1 = (const float*)d_in[1];
  const float* W2 = (const float*)d_in[2];
  float* out = (float*)d_out;

  // 1024 M-tiles, 4 tiles x 2 K-slices per 256-thread block (8 waves)
  dim3 grid(256), block(256);
  hipLaunchKernelGGL(ddb_fused_kernel, grid, block, LDS_TOTAL, stream,
                     x, W1, W2, out);
}